// HybridConv_25709674234348
// MI455X (gfx1250) — compile-verified
//
#include <hip/hip_runtime.h>
#include <stdint.h>

#define N_NODES 100000
#define N_EDGES 1000000
#define C 64
#define APAD 260   // 16-row A panel stride; (4*row + j) mod 64 banks -> conflict-free

typedef float v2f __attribute__((ext_vector_type(2)));
typedef float v8f __attribute__((ext_vector_type(8)));

// Order-preserving monotonic encoding of f32 into u32 so float-max can use
// a single hardware global_atomic_max_u32 (no CAS loop).
__device__ __forceinline__ unsigned enc_f32(float f) {
    unsigned u = __float_as_uint(f);
    return (u & 0x80000000u) ? ~u : (u | 0x80000000u);
}
__device__ __forceinline__ float dec_f32(unsigned e) {
    unsigned u = (e & 0x80000000u) ? (e ^ 0x80000000u) : ~e;
    return __uint_as_float(u);
}

__global__ void zero_ws_kernel(float* __restrict__ ws, long n) {
    long i = (long)blockIdx.x * blockDim.x + threadIdx.x;
    long stride = (long)gridDim.x * blockDim.x;
    for (; i < n; i += stride) ws[i] = 0.0f;
}

// Fused weight in K-pair-interleaved layout:
//   logical Wbig[256][64] = [W_l[0]; W_l[1]; W_l[2]; W_r[0]+W_r[1]+W_r[2]]
//   WbigP[p][col] = float2{ Wbig[2p][col], Wbig[2p+1][col] },  p = 0..127
// so a lane's B fragment for one WMMA is a single aligned 8-byte load.
__device__ __forceinline__ float wbig_row(const float* Wl, const float* Wr,
                                          int j, int d) {
    if (j < 192) return Wl[j * 64 + d];
    int r = j - 192;
    return Wr[r * 64 + d] + Wr[4096 + r * 64 + d] + Wr[8192 + r * 64 + d];
}

__global__ void prep_w_kernel(const float* __restrict__ Wl,
                              const float* __restrict__ Wr,
                              const float* __restrict__ b,
                              float2* __restrict__ WbigP,
                              float* __restrict__ bsum) {
    int idx = blockIdx.x * blockDim.x + threadIdx.x;   // 0..8191
    int p = idx >> 6;
    int d = idx & 63;
    float2 v;
    v.x = wbig_row(Wl, Wr, 2 * p + 0, d);
    v.y = wbig_row(Wl, Wr, 2 * p + 1, d);
    WbigP[idx] = v;
    if (idx < 64) bsum[idx] = b[idx] + b[64 + idx] + b[128 + idx];
}

// 16 threads per edge, 4 channels each (float4 gather), atomics into L2-resident
// accumulators. seg_max uses the monotonic-u32 encoding.
__global__ void scatter_edges_kernel(const float* __restrict__ x,
                                     const int* __restrict__ src,
                                     const int* __restrict__ dst,
                                     float* __restrict__ seg_sum,
                                     unsigned* __restrict__ seg_max,
                                     unsigned* __restrict__ cnt) {
    long tid    = (long)blockIdx.x * blockDim.x + threadIdx.x;
    long total  = (long)N_EDGES * 16;
    long stride = (long)gridDim.x * blockDim.x;
    for (; tid < total; tid += stride) {
        int e  = (int)(tid >> 4);
        int c0 = ((int)tid & 15) << 2;
        int s = src[e];
        int d = dst[e];
        const float4 v = *(const float4*)(x + (long)s * C + c0);
        float*    ss = seg_sum + (long)d * C + c0;
        unsigned* sm = seg_max + (long)d * C + c0;
        atomicAdd(ss + 0, v.x);
        atomicAdd(ss + 1, v.y);
        atomicAdd(ss + 2, v.z);
        atomicAdd(ss + 3, v.w);
        atomicMax(sm + 0, enc_f32(v.x));
        atomicMax(sm + 1, enc_f32(v.y));
        atomicMax(sm + 2, enc_f32(v.z));
        atomicMax(sm + 3, enc_f32(v.w));
        if ((tid & 15) == 0) atomicAdd(cnt + d, 1u);
    }
}

// One block = 16 nodes. Build A=[sum|mean|max|x] panel (16x256) in LDS, then
// 4 waves each compute one 16x16 output tile via fp32 WMMA over K=256
// (64 fully-unrolled v_wmma_f32_16x16x4_f32; A via ds_load_2addr_b64,
// B via one global_load_b64 per WMMA from the pair-interleaved weight).
__global__ __launch_bounds__(128) void combine_wmma_kernel(
        const float*    __restrict__ x,
        const float*    __restrict__ seg_sum,
        const unsigned* __restrict__ seg_max,
        const unsigned* __restrict__ cnt,
        const float2*   __restrict__ WbigP,
        const float*    __restrict__ bsum,
        float*          __restrict__ out) {
    __shared__ float    Ap[16 * APAD];
    __shared__ float    s_invc[16];
    __shared__ unsigned s_cnt[16];

    const int  t    = threadIdx.x;
    const long base = (long)blockIdx.x * 16;

    if (t < 16) {
        unsigned cn = cnt[base + t];
        s_cnt[t]  = cn;
        s_invc[t] = 1.0f / fmaxf((float)cn, 1.0f);
    }
    __syncthreads();

    // Fill the 16x256 A panel cooperatively.
    for (int idx = t; idx < 16 * 256; idx += 128) {
        int  row = idx >> 8;
        int  j   = idx & 255;
        long n   = base + row;
        float v;
        if (j < 64) {
            v = seg_sum[n * 64 + j];
        } else if (j < 128) {
            v = seg_sum[n * 64 + (j - 64)] * s_invc[row];
        } else if (j < 192) {
            v = (s_cnt[row] != 0u) ? dec_f32(seg_max[n * 64 + (j - 128)]) : 0.0f;
        } else {
            v = x[n * 64 + (j - 192)];
        }
        Ap[row * APAD + j] = v;
    }
    __syncthreads();

    const int lane = t & 31;
    const int wave = t >> 5;          // column tile 0..3
    const int nl   = lane & 15;
    const int half = lane >> 4;
    const int col  = wave * 16 + nl;  // output column 0..63
    const int m    = nl;              // A-matrix row held by this lane

    // C init: bias (same value for every row of the tile column).
    v8f acc;
    const float bv = bsum[col];
    #pragma unroll
    for (int r = 0; r < 8; ++r) acc[r] = bv;

    const float*  Arow = Ap + m * APAD;
    const float2* Bcol = WbigP + col;       // stride 64 float2 per K-pair

    #pragma unroll
    for (int k = 0; k < 256; k += 4) {
        v2f a, bb;
        // A 16x4: VGPR0 = K{0,2}, VGPR1 = K{1,3} split by lane half
        a.x = Arow[k + 2 * half + 0];
        a.y = Arow[k + 2 * half + 1];
        // B 4x16: one 8B load fetches both K rows of this lane's pair
        const float2 bw = Bcol[((k >> 1) + half) * 64];
        bb.x = bw.x;
        bb.y = bw.y;
        acc = __builtin_amdgcn_wmma_f32_16x16x4_f32(
                  false, a, false, bb, (short)0, acc, false, false);
    }

    // D layout: VGPR r holds M = r + 8*half, N = lane&15.
    #pragma unroll
    for (int r = 0; r < 8; ++r) {
        long n = base + r + 8 * half;
        out[n * 64 + col] = acc[r];
    }
}

extern "C" void kernel_launch(void* const* d_in, const int* in_sizes, int n_in,
                              void* d_out, int out_size, void* d_ws, size_t ws_size,
                              hipStream_t stream) {
    (void)in_sizes; (void)n_in; (void)out_size; (void)ws_size;

    const float* x  = (const float*)d_in[0];
    const int*   ei = (const int*)d_in[1];
    const float* Wl = (const float*)d_in[2];
    const float* Wr = (const float*)d_in[3];
    const float* b  = (const float*)d_in[4];
    float*       out = (float*)d_out;

    float*    ws      = (float*)d_ws;
    float*    seg_sum = ws;                                         // N*64 f32
    unsigned* seg_max = (unsigned*)(ws + (long)N_NODES * 64);       // N*64 u32
    unsigned* cnt     = (unsigned*)(ws + (long)N_NODES * 128);      // N u32
    float2*   WbigP   = (float2*)(ws + (long)N_NODES * 128 + N_NODES); // 128*64 float2
    float*    bsum    = (float*)(WbigP + 128 * 64);                 // 64 f32

    const int* src = ei;
    const int* dst = ei + N_EDGES;

    const long zero_n = (long)N_NODES * 128 + N_NODES;
    zero_ws_kernel<<<2048, 256, 0, stream>>>(ws, zero_n);
    prep_w_kernel<<<32, 256, 0, stream>>>(Wl, Wr, b, WbigP, bsum);
    scatter_edges_kernel<<<62500, 256, 0, stream>>>(x, src, dst, seg_sum, seg_max, cnt);
    combine_wmma_kernel<<<N_NODES / 16, 128, 0, stream>>>(x, seg_sum, seg_max, cnt,
                                                          WbigP, bsum, out);
}